// SymmetryLoss_58746562675301
// MI455X (gfx1250) — compile-verified
//
#include <hip/hip_runtime.h>

typedef __attribute__((ext_vector_type(2))) float v2f;
typedef __attribute__((ext_vector_type(8))) float v8f;

#define TILE 16
#define MAIN_BLOCKS 2048
#define MAIN_THREADS 256

// ---------------------------------------------------------------------------
// Setup: build the 18x4 channel table chan[ch] = [M_row(3) | c] for the 6
// affine transforms (3 reflections, 3 un-normalized quaternion rotations).
// ---------------------------------------------------------------------------
__global__ void symm_setup_kernel(const float* __restrict__ planes,
                                  const float* __restrict__ axes,
                                  float* __restrict__ chan /* 18*4 floats */) {
  if (threadIdx.x != 0 || blockIdx.x != 0) return;
  // Reflections: p' = (I - s n n^T) p - s d n,  s = 2/|n|^2
  for (int t = 0; t < 3; ++t) {
    float nx = planes[4*t+0], ny = planes[4*t+1], nz = planes[4*t+2], d = planes[4*t+3];
    float nn = nx*nx + ny*ny + nz*nz;
    float s  = 2.0f / nn;
    float n3[3] = {nx, ny, nz};
    for (int r = 0; r < 3; ++r) {
      float* row = chan + 4*(3*t + r);
      row[0] = (r==0 ? 1.0f : 0.0f) - s*n3[r]*nx;
      row[1] = (r==1 ? 1.0f : 0.0f) - s*n3[r]*ny;
      row[2] = (r==2 ? 1.0f : 0.0f) - s*n3[r]*nz;
      row[3] = -s * d * n3[r];
    }
  }
  // Rotations: q p q_conj for arbitrary (un-normalized) q = (w,x,y,z)
  for (int t = 0; t < 3; ++t) {
    float w = axes[4*t+0], x = axes[4*t+1], y = axes[4*t+2], z = axes[4*t+3];
    float* r0 = chan + 4*(9 + 3*t + 0);
    float* r1 = chan + 4*(9 + 3*t + 1);
    float* r2 = chan + 4*(9 + 3*t + 2);
    r0[0] = w*w + x*x - y*y - z*z; r0[1] = 2.0f*(x*y - w*z);       r0[2] = 2.0f*(x*z + w*y);       r0[3] = 0.0f;
    r1[0] = 2.0f*(x*y + w*z);       r1[1] = w*w - x*x + y*y - z*z; r1[2] = 2.0f*(y*z - w*x);       r1[3] = 0.0f;
    r2[0] = 2.0f*(x*z - w*y);       r2[1] = 2.0f*(y*z + w*x);       r2[2] = w*w - x*x - y*y + z*z; r2[3] = 0.0f;
  }
}

// ---------------------------------------------------------------------------
// Per-(point,transform) contribution: grid index, gather, masked distance.
// ---------------------------------------------------------------------------
__device__ __forceinline__ float dist_term(float x, float y, float z,
                                           const float* __restrict__ cp,
                                           const int* __restrict__ vox,
                                           int R, float Rf, int Rm1) {
  int ix = (int)floorf(x * Rf); ix = ix < 0 ? 0 : (ix > Rm1 ? Rm1 : ix);
  int iy = (int)floorf(y * Rf); iy = iy < 0 ? 0 : (iy > Rm1 ? Rm1 : iy);
  int iz = (int)floorf(z * Rf); iz = iz < 0 ? 0 : (iz > Rm1 ? Rm1 : iz);
  int flat = (ix * R + iy) * R + iz;
  const float* c = cp + 3 * flat;
  float dx = x - c[0], dy = y - c[1], dz = z - c[2];
  float dist = sqrtf(dx*dx + dy*dy + dz*dz);
  return dist * (float)vox[flat];
}

// ---------------------------------------------------------------------------
// Main: each wave processes 16 points per iteration. WMMA f32 16x16x4 does
// the 18-channel affine transform on the matrix pipe (2 WMMAs / 16 points);
// VALU does index + gather + distance. Lane n handles transforms 0..2 of
// point n, lane n+16 handles transforms 3..5 of the same point.
// ---------------------------------------------------------------------------
__global__ void symm_main_kernel(const float* __restrict__ sp,   // N*3
                                 const float* __restrict__ cp,   // R^3*3
                                 const int*   __restrict__ vox,  // R^3
                                 const int*   __restrict__ resolution,
                                 const float* __restrict__ chan, // 18*4
                                 float* __restrict__ partials,
                                 long long N) {
  const int   R   = resolution[0];
  const float Rf  = (float)R;
  const int   Rm1 = R - 1;

  const int  lane = threadIdx.x & 31;
  const int  m    = lane & 15;          // A row / B column within tile
  const bool hi   = lane >= 16;

  // --- A operands (16x4 layout: lanes 0-15 hold K=0,1; lanes 16-31 K=2,3) ---
  float a1x, a1y;
  if (!hi) { a1x = chan[4*m + 0]; a1y = chan[4*m + 1]; }
  else     { a1x = chan[4*m + 2]; a1y = chan[4*m + 3]; }
  float a2x = 0.0f, a2y = 0.0f;         // rows 0,1 = channels 16,17; rest 0
  if (m < 2) {
    if (!hi) { a2x = chan[4*(16+m) + 0]; a2y = chan[4*(16+m) + 1]; }
    else     { a2x = chan[4*(16+m) + 2]; a2y = chan[4*(16+m) + 3]; }
  }
  v2f A1 = {a1x, a1y};
  v2f A2 = {a2x, a2y};

  const long long tiles      = (N + TILE - 1) / TILE;
  const int       wavesPerBlk = blockDim.x >> 5;
  const int       waveInBlk   = threadIdx.x >> 5;
  const long long waveId      = (long long)blockIdx.x * wavesPerBlk + waveInBlk;
  const long long waveStride  = (long long)gridDim.x * wavesPerBlk;

  float acc = 0.0f;
  for (long long tile = waveId; tile < tiles; tile += waveStride) {
    long long p     = tile * TILE + m;
    float     valid = (p < N) ? 1.0f : 0.0f;
    long long pc    = (p < N) ? p : 0;

    // B operand (4x16): lanes 0-15 -> (x,y); lanes 16-31 -> (z,1)
    float b0, b1;
    if (!hi) { b0 = sp[3*pc + 0]; b1 = sp[3*pc + 1]; }
    else     { b0 = sp[3*pc + 2]; b1 = 1.0f; }
    v2f B = {b0, b1};
    v8f C = {};

    // D[ch, n] = transformed channel ch of point n
    v8f D  = __builtin_amdgcn_wmma_f32_16x16x4_f32(false, A1, false, B,
                                                   (short)0, C, false, false);
    v8f D2 = __builtin_amdgcn_wmma_f32_16x16x4_f32(false, A2, false, B,
                                                   (short)0, C, false, false);

    // Exchange the 3 split channels between lane n and lane n+16.
    float ex0 = __shfl_xor(D[0],  16, 32);  // lane n   receives ch8  = t2.z
    float ex1 = __shfl_xor(D2[0], 16, 32);  // lane n+16 receives ch16 = t5.y
    float ex2 = __shfl_xor(D2[1], 16, 32);  // lane n+16 receives ch17 = t5.z

    float X0,Y0,Z0, X1,Y1,Z1, X2,Y2,Z2;
    if (!hi) {                       // transforms 0,1,2 (channels 0..8)
      X0 = D[0]; Y0 = D[1]; Z0 = D[2];
      X1 = D[3]; Y1 = D[4]; Z1 = D[5];
      X2 = D[6]; Y2 = D[7]; Z2 = ex0;
    } else {                         // transforms 3,4,5 (channels 9..17)
      X0 = D[1]; Y0 = D[2]; Z0 = D[3];
      X1 = D[4]; Y1 = D[5]; Z1 = D[6];
      X2 = D[7]; Y2 = ex1;  Z2 = ex2;
    }

    acc += valid * dist_term(X0, Y0, Z0, cp, vox, R, Rf, Rm1);
    acc += valid * dist_term(X1, Y1, Z1, cp, vox, R, Rf, Rm1);
    acc += valid * dist_term(X2, Y2, Z2, cp, vox, R, Rf, Rm1);
  }

  // Wave reduction (wave32), then block reduction.
  for (int off = 16; off > 0; off >>= 1) acc += __shfl_xor(acc, off, 32);
  __shared__ float sred[32];
  if (lane == 0) sred[waveInBlk] = acc;
  __syncthreads();
  if (threadIdx.x == 0) {
    float s = 0.0f;
    for (int i = 0; i < wavesPerBlk; ++i) s += sred[i];
    partials[blockIdx.x] = s;
  }
}

// ---------------------------------------------------------------------------
// Final reduction: sum block partials, divide by N.
// ---------------------------------------------------------------------------
__global__ void symm_reduce_kernel(const float* __restrict__ partials, int n,
                                   float* __restrict__ out, float invN) {
  __shared__ float sred[256];
  float s = 0.0f;
  for (int i = threadIdx.x; i < n; i += blockDim.x) s += partials[i];
  sred[threadIdx.x] = s;
  __syncthreads();
  for (int off = 128; off > 0; off >>= 1) {
    if ((int)threadIdx.x < off) sred[threadIdx.x] += sred[threadIdx.x + off];
    __syncthreads();
  }
  if (threadIdx.x == 0) out[0] = sred[0] * invN;
}

extern "C" void kernel_launch(void* const* d_in, const int* in_sizes, int n_in,
                              void* d_out, int out_size, void* d_ws, size_t ws_size,
                              hipStream_t stream) {
  const float* sp         = (const float*)d_in[0]; // sample_points  (N*3)
  const float* cp         = (const float*)d_in[1]; // closest_points (R^3*3)
  const int*   vox        = (const int*)  d_in[2]; // voxels         (R^3)
  const float* planes     = (const float*)d_in[3]; // (3*4)
  const float* axes       = (const float*)d_in[4]; // (3*4)
  const int*   resolution = (const int*)  d_in[5]; // scalar

  long long N = (long long)in_sizes[0] / 3;

  float* chan     = (float*)d_ws;   // 72 floats used, pad to 128
  float* partials = chan + 128;     // MAIN_BLOCKS floats

  symm_setup_kernel<<<1, 32, 0, stream>>>(planes, axes, chan);
  symm_main_kernel<<<MAIN_BLOCKS, MAIN_THREADS, 0, stream>>>(
      sp, cp, vox, resolution, chan, partials, N);
  symm_reduce_kernel<<<1, 256, 0, stream>>>(partials, MAIN_BLOCKS,
                                            (float*)d_out, 1.0f / (float)N);
}